// DiffParticleFilter_88252987998713
// MI455X (gfx1250) — compile-verified
//
#include <hip/hip_runtime.h>

#define PF_N 64
#define PF_M 131072
#define PF_MLOG 17
#define PF_D 4
#define PF_ALPHA 0.5f
#define PF_BETA ((1.0f - PF_ALPHA) / (float)PF_M)

typedef float v2f __attribute__((ext_vector_type(2)));
typedef float v8f __attribute__((ext_vector_type(8)));

// ---------------------------------------------------------------------------
// Kernel 1: per-row inclusive scan of probs = ALPHA*w + (1-ALPHA)/M  -> cdf
// One block (1024 threads = 32 wave32) per row; 32 chunks of 4096 elements.
// ---------------------------------------------------------------------------
__global__ __launch_bounds__(1024) void k_scan(const float* __restrict__ weights,
                                               float* __restrict__ cdf) {
  const int n = blockIdx.x;
  const int t = threadIdx.x;
  __shared__ float sh[1024];
  const float* wrow = weights + (size_t)n * PF_M;
  float* crow = cdf + (size_t)n * PF_M;
  float run = 0.0f;
  for (int base = 0; base < PF_M; base += 4096) {
    const int i0 = base + t * 4;
    if (base + 4096 < PF_M) {
      __builtin_prefetch(wrow + i0 + 4096, 0, 1);  // global_prefetch_b8
    }
    float4 v = *reinterpret_cast<const float4*>(wrow + i0);
    float p0 = PF_ALPHA * v.x + PF_BETA;
    float p1 = PF_ALPHA * v.y + PF_BETA;
    float p2 = PF_ALPHA * v.z + PF_BETA;
    float p3 = PF_ALPHA * v.w + PF_BETA;
    // serial inclusive prefix within the 4 elements
    float s0 = p0, s1 = s0 + p1, s2 = s1 + p2, s3 = s2 + p3;
    sh[t] = s3;
    __syncthreads();
    // Hillis-Steele inclusive scan over 1024 thread totals
    for (int off = 1; off < 1024; off <<= 1) {
      float add = (t >= off) ? sh[t - off] : 0.0f;
      __syncthreads();
      sh[t] += add;
      __syncthreads();
    }
    const float incl = sh[t];
    const float btot = sh[1023];
    const float baseOff = run + (incl - s3);
    float4 o;
    o.x = baseOff + s0; o.y = baseOff + s1; o.z = baseOff + s2; o.w = baseOff + s3;
    *reinterpret_cast<float4*>(crow + i0) = o;
    run += btot;
    __syncthreads();
  }
}

// ---------------------------------------------------------------------------
// Kernel 2: inverse-CDF search + gather + measurement update (unnormalized)
// searchsorted(cdf/T, u) == lower_bound(cdf, u*T) for T>0.
// Recomputes clamp(states+diff) at the gathered index (never materialized).
// Deterministic block partial sums of unnormalized w.
// ---------------------------------------------------------------------------
__global__ __launch_bounds__(256) void k_gather(
    const float* __restrict__ states, const float* __restrict__ weights,
    const float* __restrict__ diff, const float* __restrict__ lik,
    const float* __restrict__ u, const float* __restrict__ cdf,
    float* __restrict__ outW, float* __restrict__ outPS,
    float* __restrict__ part) {
  const int gid = blockIdx.x * 256 + threadIdx.x;
  const int n = gid >> PF_MLOG;
  const float* crow = cdf + (size_t)n * PF_M;
  const float T = crow[PF_M - 1];
  const float target = u[gid] * T;
  int lo = 0, hi = PF_M;
#pragma unroll 1
  while (lo < hi) {  // 17 iterations, L2-resident probes
    const int mid = (lo + hi) >> 1;
    const float cv = crow[mid];
    const bool c = cv < target;
    lo = c ? mid + 1 : lo;
    hi = c ? hi : mid;
  }
  const int idx = (lo < PF_M - 1) ? lo : (PF_M - 1);

  const float wi = weights[(size_t)n * PF_M + idx];
  const float pr = PF_ALPHA * wi + PF_BETA;
  const float wun = (wi / pr) * lik[gid];
  outW[gid] = wun;  // unnormalized for now

  const size_t sBase = ((size_t)n * PF_M + idx) * PF_D;
  const float4 s = *reinterpret_cast<const float4*>(states + sBase);
  const float4 d = *reinterpret_cast<const float4*>(diff + sBase);
  float4 o;
  o.x = fminf(fmaxf(s.x + d.x, -1.0f), 1.0f);
  o.y = fminf(fmaxf(s.y + d.y, -1.0f), 1.0f);
  o.z = fminf(fmaxf(s.z + d.z, -2.0f), 2.0f);
  o.w = fminf(fmaxf(s.w + d.w, -2.0f), 2.0f);
  *reinterpret_cast<float4*>(outPS + (size_t)gid * PF_D) = o;

  __shared__ float sh[256];
  sh[threadIdx.x] = wun;
  __syncthreads();
  for (int off = 128; off > 0; off >>= 1) {
    if (threadIdx.x < off) sh[threadIdx.x] += sh[threadIdx.x + off];
    __syncthreads();
  }
  if (threadIdx.x == 0) part[blockIdx.x] = sh[0];  // blockIdx.x == n*512 + b
}

// ---------------------------------------------------------------------------
// Kernel 3: deterministic row sum of 512 block partials per row
// ---------------------------------------------------------------------------
__global__ __launch_bounds__(512) void k_rowsum(const float* __restrict__ part,
                                                float* __restrict__ S) {
  __shared__ float sh[512];
  sh[threadIdx.x] = part[blockIdx.x * 512 + threadIdx.x];
  __syncthreads();
  for (int off = 256; off > 0; off >>= 1) {
    if (threadIdx.x < off) sh[threadIdx.x] += sh[threadIdx.x + off];
    __syncthreads();
  }
  if (threadIdx.x == 0) S[blockIdx.x] = sh[0];
}

// ---------------------------------------------------------------------------
// Kernel 4: normalize w
// ---------------------------------------------------------------------------
__global__ __launch_bounds__(256) void k_normalize(float* __restrict__ outW,
                                                   const float* __restrict__ S) {
  const int gid = blockIdx.x * 256 + threadIdx.x;
  const int n = gid >> PF_MLOG;
  outW[gid] = outW[gid] / S[n];
}

// ---------------------------------------------------------------------------
// Kernel 5: estimates einsum via V_WMMA_F32_16X16X4_F32.
// Per WMMA: K=4 particles. A[i][k] = w[p+k] (rows replicated),
// B[k][j] = ps[p+k][j] for j<4, else 0.  D[0][j] accumulates e_j.
// A layout (ISA 7.12.2, 32-bit A 16x4): lanes 0-15 VGPR0/1 = K0/K1,
// lanes 16-31 VGPR0/1 = K2/K3.  B symmetric (lane = column).
// B columns >= 4 are zeroed by an unconditional load from a clamped
// in-bounds address times a 0/1 mask -- no EXEC manipulation in the
// hot loop, so EXEC stays all-ones across every WMMA.
// Each wave chains 1024 WMMAs over 4096 particles; partials written
// per-wave for a deterministic final reduction.
// ---------------------------------------------------------------------------
__global__ __launch_bounds__(256) void k_est(const float* __restrict__ W,
                                             const float* __restrict__ PS,
                                             float* __restrict__ estPart) {
  const int n = blockIdx.y;
  const int wave = threadIdx.x >> 5;
  const int lane = threadIdx.x & 31;
  const int wgl = blockIdx.x * 8 + wave;   // 0..31 global wave id within row
  const int mBase = wgl * 4096;
  const float* Wr = W + (size_t)n * PF_M;
  const float* Pr = PS + (size_t)n * PF_M * PF_D;
  const int col = lane & 15;
  const int colC = col & 3;                     // clamped, always in-bounds
  const float msk = (col < PF_D) ? 1.0f : 0.0f; // zero out columns 4..15
  const int hi = (lane >> 4) & 1;
  v8f c = {0.f, 0.f, 0.f, 0.f, 0.f, 0.f, 0.f, 0.f};
#pragma unroll 4
  for (int g = 0; g < 4096; g += 4) {
    const int p = mBase + g;
    v2f a, b;
    a.x = Wr[p + 2 * hi + 0];
    a.y = Wr[p + 2 * hi + 1];
    const float* pb = Pr + (size_t)p * PF_D + 8 * hi + colC;
    b.x = pb[0] * msk;
    b.y = pb[4] * msk;
    c = __builtin_amdgcn_wmma_f32_16x16x4_f32(
        /*neg_a=*/false, a, /*neg_b=*/false, b,
        /*c_mod=*/(short)0, c, /*reuse_a=*/false, /*reuse_b=*/false);
  }
  // D rows are replicated; D[0][j] lives in VGPR0 (c[0]) of lanes j = 0..3.
  if (lane < PF_D) estPart[((size_t)(n * 32 + wgl)) * PF_D + lane] = c[0];
}

// ---------------------------------------------------------------------------
// Kernel 6: deterministic final reduction of 32 wave partials per (n, j)
// ---------------------------------------------------------------------------
__global__ __launch_bounds__(256) void k_est_final(const float* __restrict__ estPart,
                                                   float* __restrict__ est) {
  const int t = threadIdx.x;  // 256 = N*D
  const int n = t >> 2, j = t & 3;
  float s = 0.0f;
  for (int p = 0; p < 32; ++p) s += estPart[((size_t)(n * 32 + p)) * PF_D + j];
  est[t] = s;
}

// ---------------------------------------------------------------------------
extern "C" void kernel_launch(void* const* d_in, const int* in_sizes, int n_in,
                              void* d_out, int out_size, void* d_ws, size_t ws_size,
                              hipStream_t stream) {
  const float* states  = (const float*)d_in[0];
  const float* weights = (const float*)d_in[1];
  const float* diff    = (const float*)d_in[2];
  const float* lik     = (const float*)d_in[3];
  const float* u       = (const float*)d_in[4];

  float* out = (float*)d_out;
  float* est = out;                                  // [N, D]     = 256
  float* W   = out + PF_N * PF_D;                    // [N, M]
  float* PS  = W + (size_t)PF_N * PF_M;              // [N, M, D]

  // workspace layout (floats): cdf | block partials | row sums | wave partials
  float* cdf     = (float*)d_ws;
  float* part    = cdf + (size_t)PF_N * PF_M;        // N * 512
  float* S       = part + PF_N * 512;                // N
  float* estPart = S + PF_N;                         // N * 32 * D

  k_scan<<<PF_N, 1024, 0, stream>>>(weights, cdf);
  k_gather<<<(PF_N * PF_M) / 256, 256, 0, stream>>>(states, weights, diff, lik, u,
                                                    cdf, W, PS, part);
  k_rowsum<<<PF_N, 512, 0, stream>>>(part, S);
  k_normalize<<<(PF_N * PF_M) / 256, 256, 0, stream>>>(W, S);
  dim3 gEst(PF_M / 32768, PF_N);                     // (4, 64)
  k_est<<<gEst, 256, 0, stream>>>(W, PS, estPart);
  k_est_final<<<1, 256, 0, stream>>>(estPart, est);
}